// NeuralKalmanFilter_38981123178909
// MI455X (gfx1250) — compile-verified
//
#include <hip/hip_runtime.h>

// ---------------------------------------------------------------------------
// Neural Kalman Filter inference loop, MI455X (gfx1250), wave32.
// Persistent single kernel, 64 blocks x 256 threads, grid-wide spin barrier.
// fp32 WMMA (V_WMMA_F32_16X16X4_F32) for row-major matvec phases.
// ---------------------------------------------------------------------------

static constexpr int LATENT = 2048;
static constexpr int OBS    = 1024;
static constexpr int CTRL   = 512;
static constexpr int SEQ    = 200;

static constexpr int NBLK   = 64;    // persistent grid (must be co-resident)
static constexpr int NTHR   = 256;   // 8 waves (wave32)
static constexpr int NWAVE  = NTHR / 32;

typedef float v2f __attribute__((ext_vector_type(2)));
typedef float v8f __attribute__((ext_vector_type(8)));

// ---------------------------------------------------------------------------
// Grid-wide barrier: monotonic arrival counter in L2 (never reset -> no
// reset/reuse races; max value 8200*64 << 2^32). Release fence by every
// thread before arrival, acquire fence by every thread after completion
// (L0/WGP$ invalidate so post-barrier loads see other blocks' stores).
// ---------------------------------------------------------------------------
__device__ __forceinline__ void grid_barrier(unsigned* cnt, unsigned gen)
{
    __threadfence();                 // flush this thread's global writes
    __syncthreads();
    if (threadIdx.x == 0) {
        __hip_atomic_fetch_add(cnt, 1u, __ATOMIC_RELEASE, __HIP_MEMORY_SCOPE_AGENT);
        const unsigned target = gen * (unsigned)NBLK;
        while (__hip_atomic_load(cnt, __ATOMIC_ACQUIRE, __HIP_MEMORY_SCOPE_AGENT) < target) {
            __builtin_amdgcn_s_sleep(1);
        }
    }
    __syncthreads();
    __threadfence();                 // invalidate stale cached lines
}

// ---------------------------------------------------------------------------
// One wave computes a 16-row fp32 matvec partial over K-slice [k0,k1) using
// V_WMMA_F32_16X16X4_F32.  A-tile (16x4 fp32, documented layout): lane l
// holds row (l&15), K-pair base 2*(l>>4); VGPR0 = K+0, VGPR1 = K+1.
// B-tile: the 4 vector values are broadcast to every N column (lanes 0-15
// get v[k],v[k+1]; lanes 16-31 get v[k+2],v[k+3]), so all 16 D columns hold
// the identical matvec result. After the loop: lane 0's acc[r] = row r,
// lane 16's acc[r] = row 8+r.
// ---------------------------------------------------------------------------
__device__ __forceinline__ void wmma_mv16(const float* __restrict__ W, int ld,
                                          const float* __restrict__ vlds,
                                          int k0, int k1, v8f& acc)
{
    const int lane = threadIdx.x & 31;
#if __has_builtin(__builtin_amdgcn_wmma_f32_16x16x4_f32)
    const int row   = lane & 15;
    const int khalf = (lane >> 4) << 1;               // 0 or 2
    const float* wp = W + (size_t)row * (size_t)ld + khalf;
    const float* vp = vlds + khalf;
    for (int k = k0; k < k1; k += 4) {
        v2f a = *(const v2f*)(wp + k);                // global_load_b64
        v2f b = *(const v2f*)(vp + k);                // ds_load_b64 (broadcast)
        acc = __builtin_amdgcn_wmma_f32_16x16x4_f32(
            /*neg_a=*/false, a, /*neg_b=*/false, b,
            /*c_mod=*/(short)0, acc, /*reuse_a=*/false, /*reuse_b=*/false);
    }
#else
    // Scalar fallback with the same output convention (lane0 rows 0-7,
    // lane16 rows 8-15; all lanes in a half compute identical values).
    const int rbase = (lane >> 4) * 8;
    for (int k = k0; k < k1; k += 4) {
#pragma unroll
        for (int r = 0; r < 8; ++r) {
            const float* wr = W + (size_t)(rbase + r) * (size_t)ld + k;
            acc[r] += wr[0]*vlds[k] + wr[1]*vlds[k+1]
                    + wr[2]*vlds[k+2] + wr[3]*vlds[k+3];
        }
    }
#endif
}

// ---------------------------------------------------------------------------
__global__ void nkf_init(float* tzG, unsigned* cnt)
{
    const int i = blockIdx.x * blockDim.x + threadIdx.x;
    if (i < LATENT) tzG[i] = 0.0f;           // tanh(z0)=0
    if (i == 0)     *cnt = 0u;               // barrier counter
}

// ---------------------------------------------------------------------------
__global__ __launch_bounds__(NTHR, 1)
void nkf_persistent(const float* __restrict__ inputs,    // [OBS, SEQ]
                    const float* __restrict__ controls,  // [CTRL, SEQ]
                    const float* __restrict__ Wr,        // [LATENT, LATENT]
                    const float* __restrict__ Win,       // [LATENT, CTRL]
                    const float* __restrict__ Wout,      // [OBS, LATENT]
                    const float* __restrict__ lr_p,
                    const int*   __restrict__ iters_p,
                    float* __restrict__ out,             // [LATENT, SEQ]
                    float* __restrict__ tzG,             // [LATENT] tanh(z)
                    float* __restrict__ drvG,            // [LATENT]
                    float* __restrict__ exG,             // [OBS]
                    unsigned* cnt)
{
    __shared__ float vecLDS[LATENT];          // tz (phases D/A) or ex (phase B)
    __shared__ float tuLDS[CTRL];
    __shared__ float redLDS[NWAVE][16];
    __shared__ float partLDS[NTHR];

    const int tid  = threadIdx.x;
    const int bid  = blockIdx.x;
    const int wave = tid >> 5;
    const int lane = tid & 31;
    const float lr = *lr_p;
    const int iters = *iters_p;

    // Each thread tid<32 owns latent row j = bid*32 + tid; z kept in register.
    const int jown = bid * 32 + tid;          // valid when tid < 32
    float zreg  = 0.0f;
    float tzreg = 0.0f;

    unsigned gen = 0;

    for (int t = 0; t < SEQ; ++t) {
        // ---------- Phase D: drive = Wr @ tanh(z_prev) + Win @ tanh(u) -----
        for (int i = tid; i < LATENT; i += NTHR) vecLDS[i] = tzG[i];
        for (int c = tid; c < CTRL;   c += NTHR)
            tuLDS[c] = tanhf(controls[(size_t)c * SEQ + t]);
        __syncthreads();
        {
            const int g  = wave >> 2;                 // row group 0/1
            const int ws = wave & 3;                  // K-slice id
            const int j0 = bid * 32 + g * 16;
            v8f acc = {0.f,0.f,0.f,0.f,0.f,0.f,0.f,0.f};
            wmma_mv16(Wr  + (size_t)j0 * LATENT, LATENT, vecLDS,
                      ws * 512, ws * 512 + 512, acc);
            wmma_mv16(Win + (size_t)j0 * CTRL,   CTRL,   tuLDS,
                      ws * 128, ws * 128 + 128, acc);
            if (lane == 0) {
#pragma unroll
                for (int r = 0; r < 8; ++r) redLDS[wave][r] = acc[r];
            } else if (lane == 16) {
#pragma unroll
                for (int r = 0; r < 8; ++r) redLDS[wave][8 + r] = acc[r];
            }
        }
        __syncthreads();
        float drive = 0.0f;
        if (tid < 32) {                               // fixed-order reduction
            const int g = tid >> 4, m = tid & 15;
            float s = 0.0f;
            for (int w = g * 4; w < g * 4 + 4; ++w) s += redLDS[w][m];
            drive = s;
            drvG[bid * 32 + g * 16 + m] = s;          // (kept for debug/clarity)
        }
        // re-permute: thread tid<32 needs drive of row jown = bid*32+tid
        __syncthreads();
        if (tid < 32) drive = drvG[jown];
        ++gen; grid_barrier(cnt, gen);

        // x column is fixed over the inner loop
        float xown = 0.0f;
        if (tid < 16) xown = inputs[(size_t)(bid * 16 + tid) * SEQ + t];

        // ---------- Inner fixed-point loop --------------------------------
        for (int it = 0; it < iters; ++it) {
            // Phase A: ex = x - Wout @ tanh(z)   (16 obs rows per block)
            for (int i = tid; i < LATENT; i += NTHR) vecLDS[i] = tzG[i];
            __syncthreads();
            {
                const int i0 = bid * 16;
                v8f acc = {0.f,0.f,0.f,0.f,0.f,0.f,0.f,0.f};
                wmma_mv16(Wout + (size_t)i0 * LATENT, LATENT, vecLDS,
                          wave * 256, wave * 256 + 256, acc);
                if (lane == 0) {
#pragma unroll
                    for (int r = 0; r < 8; ++r) redLDS[wave][r] = acc[r];
                } else if (lane == 16) {
#pragma unroll
                    for (int r = 0; r < 8; ++r) redLDS[wave][8 + r] = acc[r];
                }
            }
            __syncthreads();
            if (tid < 16) {
                float s = 0.0f;
                for (int w = 0; w < NWAVE; ++w) s += redLDS[w][tid];
                exG[bid * 16 + tid] = xown - s;
            }
            ++gen; grid_barrier(cnt, gen);

            // Phase B: g = Wout^T @ ex (coalesced column access); z update
            for (int i = tid; i < OBS; i += NTHR) vecLDS[i] = exG[i];
            __syncthreads();
            {
                const int j  = bid * 32 + (tid & 31);     // coalesced across lanes
                const int i0 = (tid >> 5) * (OBS / NWAVE);
                float p = 0.0f;
                for (int i = i0; i < i0 + OBS / NWAVE; ++i)
                    p += Wout[(size_t)i * LATENT + j] * vecLDS[i];
                partLDS[tid] = p;
            }
            __syncthreads();
            if (tid < 32) {
                float gsum = 0.0f;
                for (int s = 0; s < NWAVE; ++s) gsum += partLDS[s * 32 + tid];
                const float zv  = zreg;
                const float ez  = zv - drive;
                const float sgn = (zv > 0.0f) ? 1.0f : ((zv < 0.0f) ? -1.0f : 0.0f);
                const float dz  = ez - (1.0f - tzreg * tzreg) * gsum + 0.5f * sgn;
                const float zn  = zv - lr * dz;
                zreg  = zn;
                tzreg = tanhf(zn);
                tzG[jown] = tzreg;                    // visible to all blocks
            }
            ++gen; grid_barrier(cnt, gen);
        }

        // ---------- emit zs[:, t] -----------------------------------------
        if (tid < 32) out[(size_t)jown * SEQ + t] = zreg;
    }
}

// ---------------------------------------------------------------------------
extern "C" void kernel_launch(void* const* d_in, const int* in_sizes, int n_in,
                              void* d_out, int out_size, void* d_ws, size_t ws_size,
                              hipStream_t stream)
{
    (void)in_sizes; (void)n_in; (void)out_size; (void)ws_size;

    const float* inputs   = (const float*)d_in[0];
    const float* controls = (const float*)d_in[1];
    const float* Wr       = (const float*)d_in[2];
    const float* Win      = (const float*)d_in[3];
    const float* Wout     = (const float*)d_in[4];
    const float* lr       = (const float*)d_in[5];
    const int*   iters    = (const int*)d_in[6];
    float* out = (float*)d_out;

    float* tzG  = (float*)d_ws;              // [LATENT]
    float* drvG = tzG + LATENT;              // [LATENT]
    float* exG  = drvG + LATENT;             // [OBS]
    unsigned* cnt = (unsigned*)(exG + OBS);  // barrier counter

    nkf_init<<<(LATENT + NTHR - 1) / NTHR, NTHR, 0, stream>>>(tzG, cnt);
    nkf_persistent<<<NBLK, NTHR, 0, stream>>>(inputs, controls, Wr, Win, Wout,
                                              lr, iters, out,
                                              tzG, drvG, exG, cnt);
}